// ConvolutionFromEdgeSetUpdate_21852793602279
// MI455X (gfx1250) — compile-verified
//
#include <hip/hip_runtime.h>

// GNN edge update + segment-sum for MI455X (gfx1250, wave32).
//   messages = relu(concat(node[src], node[tgt], edge) @ W + b)   [800k x 64]
//   out[t]  += messages[e] for tgt_idx[e] == t                    [50k  x 64]
// Memory-bound (~1 GB moved vs 19.7 GFLOP) -> keep f32 precision and use
// V_WMMA_F32_16X16X4_F32 for the 192-deep GEMM.
// W is staged TRANSPOSED in LDS so both A and B fragments are single
// contiguous ds_load_b64 reads (no VGPR-pair assembly moves).

#define D_FEAT 64
#define UNITS 64
#define K_DIM 192                   // 3 * D_FEAT
#define EDGES_PER_WAVE 16
#define WAVES_PER_BLOCK 4
#define EDGES_PER_BLOCK (EDGES_PER_WAVE * WAVES_PER_BLOCK)  // 64
// stride 196 dwords: 196 % 64 == 4 -> lanes 0-15 (rows m, col kb) hit bank
// pairs {4m, 4m+1}, lanes 16-31 (col kb+2) hit {4m+2, 4m+3}: conflict-free b64.
#define MSG_STRIDE 196
#define WT_STRIDE 196

typedef __attribute__((ext_vector_type(2))) float v2f;
typedef __attribute__((ext_vector_type(8))) float v8f;

__global__ __launch_bounds__(128) void edge_update_scatter(
    const float* __restrict__ node_feat,
    const float* __restrict__ edge_feat,
    const int*   __restrict__ src_idx,
    const int*   __restrict__ tgt_idx,
    const float* __restrict__ W,      // [192][64] row-major in global
    const float* __restrict__ bias,   // [64]
    float* __restrict__ out,          // [n_nodes][64], pre-zeroed
    int n_edges)
{
    __shared__ float ldsWt[UNITS * WT_STRIDE];                               // 49 KB, Wt[n][k]
    __shared__ float ldsMsg[WAVES_PER_BLOCK * EDGES_PER_WAVE * MSG_STRIDE];  // 49 KB

    const int tid  = threadIdx.x;
    const int lane = tid & 31;
    const int wave = tid >> 5;

    // ---- stage W transposed into LDS: Wt[n][k] = W[k][n] ----
    // Coalesced float4 global reads; 4x b32 transpose-scatter stores (one-time).
    for (int i = tid; i < K_DIM * (D_FEAT / 4); i += blockDim.x) {  // 3072 float4
        const int k  = i >> 4;      // 16 float4 per 64-float W row
        const int c4 = i & 15;
        const float4 w4 = ((const float4*)W)[k * 16 + c4];
        const int n = c4 * 4;
        ldsWt[(n + 0) * WT_STRIDE + k] = w4.x;
        ldsWt[(n + 1) * WT_STRIDE + k] = w4.y;
        ldsWt[(n + 2) * WT_STRIDE + k] = w4.z;
        ldsWt[(n + 3) * WT_STRIDE + k] = w4.w;
    }

    // ---- gather this wave's 16 edges into an LDS 16x192 message tile ----
    const int edge_base = blockIdx.x * EDGES_PER_BLOCK + wave * EDGES_PER_WAVE;
    float* msg = &ldsMsg[wave * EDGES_PER_WAVE * MSG_STRIDE];

    for (int r = 0; r < EDGES_PER_WAVE; ++r) {
        const int e  = edge_base + r;
        const int ec = (e < n_edges) ? e : (n_edges - 1);   // clamp for tail safety
        const int s  = src_idx[ec];
        const int t  = tgt_idx[ec];
        const float4* p0 = (const float4*)(node_feat + (size_t)s  * D_FEAT);
        const float4* p1 = (const float4*)(node_feat + (size_t)t  * D_FEAT);
        const float4* p2 = (const float4*)(edge_feat + (size_t)ec * D_FEAT);
        for (int c = lane; c < K_DIM / 4; c += 32) {         // 48 float4 per row
            const int seg = c >> 4;                          // 0: src, 1: tgt, 2: edge
            const int off = c & 15;
            const float4* p = (seg == 0) ? p0 : ((seg == 1) ? p1 : p2);
            *(float4*)&msg[r * MSG_STRIDE + c * 4] = p[off];
        }
    }
    __syncthreads();

    // ---- GEMM: 16 edges x 192 x 64 via V_WMMA_F32_16X16X4_F32 ----
    // A 16x4 f32 layout: lane = 16*(K/2) + M, VGPR j holds K = 2*(lane>>4) + j.
    // B 4x16 mirrored:   lane = 16*(K/2) + N.
    // Both fragments are one contiguous ds_load_b64 (K pair is adjacent in
    // msg rows and in transposed Wt rows).
    const int m  = lane & 15;
    const int kh = lane >> 4;       // 0 or 1

    const float* arow = &msg[m * MSG_STRIDE + 2 * kh];
    const float* brow0 = &ldsWt[( 0 + m) * WT_STRIDE + 2 * kh];
    const float* brow1 = &ldsWt[(16 + m) * WT_STRIDE + 2 * kh];
    const float* brow2 = &ldsWt[(32 + m) * WT_STRIDE + 2 * kh];
    const float* brow3 = &ldsWt[(48 + m) * WT_STRIDE + 2 * kh];

    v8f acc0 = {}, acc1 = {}, acc2 = {}, acc3 = {};

    for (int kb = 0; kb < K_DIM; kb += 4) {
        const v2f a  = *(const v2f*)(arow  + kb);
        const v2f b0 = *(const v2f*)(brow0 + kb);
        const v2f b1 = *(const v2f*)(brow1 + kb);
        const v2f b2 = *(const v2f*)(brow2 + kb);
        const v2f b3 = *(const v2f*)(brow3 + kb);
        acc0 = __builtin_amdgcn_wmma_f32_16x16x4_f32(false, a, false, b0, (short)0, acc0, false, false);
        acc1 = __builtin_amdgcn_wmma_f32_16x16x4_f32(false, a, false, b1, (short)0, acc1, false, false);
        acc2 = __builtin_amdgcn_wmma_f32_16x16x4_f32(false, a, false, b2, (short)0, acc2, false, false);
        acc3 = __builtin_amdgcn_wmma_f32_16x16x4_f32(false, a, false, b3, (short)0, acc3, false, false);
    }

    // ---- bias + ReLU + scatter-add (segment sum) ----
    // C/D layout: VGPR r holds row M = r + 8*kh, col N = m (within 16-wide tile).
    const float bv0 = bias[ 0 + m];
    const float bv1 = bias[16 + m];
    const float bv2 = bias[32 + m];
    const float bv3 = bias[48 + m];

    for (int r = 0; r < 8; ++r) {
        const int row = r + 8 * kh;
        const int e   = edge_base + row;
        if (e < n_edges) {
            const int tg = tgt_idx[e];
            float* orow = out + (size_t)tg * UNITS;
            float v0 = acc0[r] + bv0; v0 = v0 > 0.f ? v0 : 0.f;
            float v1 = acc1[r] + bv1; v1 = v1 > 0.f ? v1 : 0.f;
            float v2 = acc2[r] + bv2; v2 = v2 > 0.f ? v2 : 0.f;
            float v3 = acc3[r] + bv3; v3 = v3 > 0.f ? v3 : 0.f;
            unsafeAtomicAdd(&orow[ 0 + m], v0);   // global_atomic_add_f32
            unsafeAtomicAdd(&orow[16 + m], v1);
            unsafeAtomicAdd(&orow[32 + m], v2);
            unsafeAtomicAdd(&orow[48 + m], v3);
        }
    }
}

extern "C" void kernel_launch(void* const* d_in, const int* in_sizes, int n_in,
                              void* d_out, int out_size, void* d_ws, size_t ws_size,
                              hipStream_t stream) {
    const float* node_feat = (const float*)d_in[0];
    const float* edge_feat = (const float*)d_in[1];
    const int*   src_idx   = (const int*)d_in[2];
    const int*   tgt_idx   = (const int*)d_in[3];
    const float* W         = (const float*)d_in[4];
    const float* bias      = (const float*)d_in[5];
    float* out = (float*)d_out;

    const int n_edges = in_sizes[2];

    // Zero the segment-sum accumulator (graph-capture-safe async memset).
    hipMemsetAsync(d_out, 0, (size_t)out_size * sizeof(float), stream);

    const int grid = (n_edges + EDGES_PER_BLOCK - 1) / EDGES_PER_BLOCK;
    edge_update_scatter<<<grid, 128, 0, stream>>>(
        node_feat, edge_feat, src_idx, tgt_idx, W, bias, out, n_edges);
}